// GraphSAGELayer_39831526703907
// MI455X (gfx1250) — compile-verified
//
#include <hip/hip_runtime.h>

#define FEAT 128  // D in the reference

typedef float v2f __attribute__((ext_vector_type(2)));
typedef float v8f __attribute__((ext_vector_type(8)));

// Native CDNA5 no-return f32 atomic add (tracked with STOREcnt, handled in
// the L2 atomic units).  Inline asm guarantees global_atomic_add_f32 is
// emitted instead of a CAS loop fallback.
__device__ __forceinline__ void atomic_fadd_l2(float* p, float v) {
    asm volatile("global_atomic_add_f32 %0, %1, off"
                 :
                 : "v"((unsigned long long)(uintptr_t)p), "v"(v)
                 : "memory");
}

// ---------------------------------------------------------------------------
// Kernel 1: zero agg[N*D] and deg[N] in workspace (harness poisons ws to 0xAA)
// ---------------------------------------------------------------------------
__global__ __launch_bounds__(256) void sage_zero_ws(float* __restrict__ p, long n) {
    long i = (long)blockIdx.x * blockDim.x + threadIdx.x;
    long stride = (long)gridDim.x * blockDim.x;
    for (; i < n; i += stride) p[i] = 0.0f;
}

// ---------------------------------------------------------------------------
// Kernel 2: edge scatter.  One wave32 per edge; lane L handles 4 floats
// [L*4 .. L*4+3] of the 128-float feature row.  x and agg (51 MB each) are
// L2-resident (192 MB L2), so the f32 atomics run in the L2 atomic units.
// ---------------------------------------------------------------------------
__global__ __launch_bounds__(256) void sage_scatter(const float* __restrict__ x,
                                                    const int*   __restrict__ src,
                                                    const int*   __restrict__ dst,
                                                    float* __restrict__ agg,
                                                    float* __restrict__ deg,
                                                    int nEdges) {
    long t   = (long)blockIdx.x * blockDim.x + threadIdx.x;
    int lane = (int)(t & 31);
    long e   = t >> 5;
    if (e >= nEdges) return;

    int s = src[e];
    int d = dst[e];

    const float4 v = *(const float4*)(x + (long)s * FEAT + lane * 4);
    float* a = agg + (long)d * FEAT + lane * 4;

    atomic_fadd_l2(a + 0, v.x);
    atomic_fadd_l2(a + 1, v.y);
    atomic_fadd_l2(a + 2, v.z);
    atomic_fadd_l2(a + 3, v.w);
    if (lane == 0) atomic_fadd_l2(deg + d, 1.0f);
}

// ---------------------------------------------------------------------------
// K-reduction pass: acc[j] += (scale? a*inv : a) . B_j  over K=128 in steps
// of 4 (V_WMMA_F32_16X16X4_F32).  One WMMA group of 8 per iteration so the
// accumulator chain register-coalesces (no v_mov shuffles); A fragments are
// software-pipelined one kt ahead; B fragments batch-loaded from LDS.
// ---------------------------------------------------------------------------
template <bool SCALE>
__device__ __forceinline__ void sage_kpass(const float* __restrict__ arow,
                                           const float* __restrict__ lb,
                                           float inv, int koff, int sw,
                                           v8f (&acc)[8]) {
    v2f a = *(const v2f*)(arow + koff);
    for (int kt = 0; kt < 32; ++kt) {
        const int ks = (kt * 4 + koff) ^ sw;        // swizzled k within col

        v2f an = a;                                  // defined on last iter too
        if (kt < 31) an = *(const v2f*)(arow + (kt + 1) * 4 + koff);

        v2f b[8];
        #pragma unroll
        for (int j = 0; j < 8; ++j)
            b[j] = *(const v2f*)(lb + (j << 11) + ks);   // j*16 cols * 128

        v2f av = a;
        if (SCALE) { av.x *= inv; av.y *= inv; }

        #pragma unroll
        for (int j = 0; j < 8; ++j)
            acc[j] = __builtin_amdgcn_wmma_f32_16x16x4_f32(
                false, av, false, b[j], (short)0, acc[j], false, false);

        a = an;
    }
}

// ---------------------------------------------------------------------------
// Kernel 3: fused  out = relu(x @ Wself^T + (agg/deg) @ Wneigh^T + bs + bn)
// Block = 8 waves = 8 m-tiles.  Both weight matrices staged in LDS (128 KB)
// with an XOR bank swizzle (k ^ ((col&15)<<2)) -> conflict-free ds_load_b64.
// Two sequential K passes (x·Wself, then agg·Wneigh) over shared accumulators.
// ---------------------------------------------------------------------------
__global__ __launch_bounds__(256) void sage_gemm(const float* __restrict__ x,
                                                 const float* __restrict__ agg,
                                                 const float* __restrict__ deg,
                                                 const float* __restrict__ Wself,
                                                 const float* __restrict__ bself,
                                                 const float* __restrict__ Wneigh,
                                                 const float* __restrict__ bneigh,
                                                 float* __restrict__ out,
                                                 int nTiles, int nNodes) {
    __shared__ float wlds[2 * FEAT * FEAT];   // 128 KB: [Wself | Wneigh], swizzled

    const int tid = threadIdx.x;

    // ---- cooperative staging of both weight matrices into LDS (swizzled) ----
    {
        const float* Wm[2] = { Wself, Wneigh };
        #pragma unroll
        for (int m = 0; m < 2; ++m) {
            float* dl = wlds + m * (FEAT * FEAT);
            for (int i = tid * 4; i < FEAT * FEAT; i += 256 * 4) {
                float4 v = *(const float4*)(Wm[m] + i);
                const int col = i >> 7;          // row of W = output column j
                const int k   = i & 127;         // multiple of 4
                const int idx = (col << 7) + (k ^ ((col & 15) << 2));
                *(float4*)(dl + idx) = v;
            }
        }
    }
    __syncthreads();

    const int lane = tid & 31;
    const int mt   = blockIdx.x * 8 + (tid >> 5);   // m-tile for this wave
    if (mt >= nTiles) return;

    const int lrow = lane & 15;          // A row / B,C col within tile
    const int half = lane >> 4;          // K-pair select / C row-half select
    const int koff = half << 1;          // 0 or 2
    const int sw   = lrow << 2;          // per-lane XOR swizzle constant

    int rowA = (mt << 4) + lrow;
    if (rowA >= nNodes) rowA = nNodes - 1;   // tail clamp (stores are guarded)

    const float inv = 1.0f / fmaxf(deg[rowA], 1.0f);

    const float* xrow = x   + (long)rowA * FEAT;
    const float* arow = agg + (long)rowA * FEAT;
    const float* lbs  = wlds + lrow * FEAT;                 // Wself block base
    const float* lbn  = wlds + FEAT * FEAT + lrow * FEAT;   // Wneigh block base

    v8f acc[8] = {};

    // pass 1: x @ Wself^T ; pass 2: (agg * inv) @ Wneigh^T
    sage_kpass<false>(xrow, lbs, 1.0f, koff, sw, acc);
    sage_kpass<true >(arow, lbn, inv,  koff, sw, acc);

    // ---- epilogue: bias + ReLU + store.  C layout: VGPR v -> row v + half*8 ----
    const int mbase = (mt << 4) + half * 8;
    #pragma unroll
    for (int j = 0; j < 8; ++j) {
        const int col  = (j << 4) + lrow;
        const float bias = bself[col] + bneigh[col];
        #pragma unroll
        for (int v = 0; v < 8; ++v) {
            const int m = mbase + v;
            if (m < nNodes) {
                float val = acc[j][v] + bias;
                out[(long)m * FEAT + col] = fmaxf(val, 0.0f);
            }
        }
    }
}

// ---------------------------------------------------------------------------
extern "C" void kernel_launch(void* const* d_in, const int* in_sizes, int n_in,
                              void* d_out, int out_size, void* d_ws, size_t ws_size,
                              hipStream_t stream) {
    const float* x      = (const float*)d_in[0];
    const int*   edges  = (const int*)  d_in[1];   // [2, E] int32 (JAX default x32)
    const float* Wself  = (const float*)d_in[2];
    const float* bself  = (const float*)d_in[3];
    const float* Wneigh = (const float*)d_in[4];
    const float* bneigh = (const float*)d_in[5];
    float*       out    = (float*)d_out;

    const int nNodes = in_sizes[0] / FEAT;
    const int nEdges = in_sizes[1] / 2;

    float* agg = (float*)d_ws;                       // [N, 128]
    float* deg = agg + (size_t)nNodes * FEAT;        // [N]

    // 1) zero workspace accumulators
    const long zcount = (long)nNodes * FEAT + nNodes;
    sage_zero_ws<<<2048, 256, 0, stream>>>(agg, zcount);

    // 2) scatter-sum over edges: one wave per edge
    const long sthreads = (long)nEdges * 32;
    const int  sblocks  = (int)((sthreads + 255) / 256);
    sage_scatter<<<sblocks, 256, 0, stream>>>(x, edges, edges + nEdges,
                                              agg, deg, nEdges);

    // 3) fused normalize + dual WMMA GEMM + bias + ReLU (LDS-staged weights)
    const int nTiles  = (nNodes + 15) / 16;
    const int gblocks = (nTiles + 7) / 8;
    sage_gemm<<<gblocks, 256, 0, stream>>>(x, agg, deg, Wself, bself,
                                           Wneigh, bneigh, out, nTiles, nNodes);
}